// Expression_82643760709953
// MI455X (gfx1250) — compile-verified
//
#include <hip/hip_runtime.h>
#include <math.h>

// Elementwise map over 8,388,608 rows x 8 fp32 vars:
//   out = tanh( (x0*x1 + sin(x2)) * exp(-|x3|) + x4 / (x5*x5 + exp(x6)) - x7 )
//
// Strictly HBM-bound on MI455X: 288 MB streamed @ 23.3 TB/s => ~12.4 us floor.
// Strategy: 4 rows per thread, b128 non-temporal loads (2 per row, rows are
// 32B contiguous), b128 non-temporal store of 4 results. All transcendentals
// on the CDNA5 TRANS pipe: v_sin_f32, v_exp_f32, v_rcp_f32, v_tanh_f32
// (gfx1250 hardware tanh) — fully hidden under memory time.

// Clang native vector type: required by __builtin_nontemporal_load/store.
typedef float v4f __attribute__((ext_vector_type(4)));

__device__ __forceinline__ float fast_tanh(float x) {
#if __has_builtin(__builtin_amdgcn_tanhf)
    return __builtin_amdgcn_tanhf(x);      // v_tanh_f32 (gfx1250)
#else
    float r;
    asm("v_tanh_f32 %0, %1" : "=v"(r) : "v"(x));
    return r;
#endif
}

__device__ __forceinline__ float fast_rcp(float x) {
#if __has_builtin(__builtin_amdgcn_rcpf)
    return __builtin_amdgcn_rcpf(x);       // v_rcp_f32 (~1 ulp; den > 0 here)
#else
    float r;
    asm("v_rcp_f32 %0, %1" : "=v"(r) : "v"(x));
    return r;
#endif
}

__device__ __forceinline__ float eval_row(const v4f a, const v4f b) {
    // a = {x0,x1,x2,x3}, b = {x4,x5,x6,x7}
    float t1  = fmaf(a.x, a.y, __sinf(a.z));         // x0*x1 + sin(x2)
    float t2  = __expf(-fabsf(a.w));                 // exp(-|x3|)
    float den = fmaf(b.y, b.y, __expf(b.z));         // x5*x5 + exp(x6), > 0
    float e   = fmaf(t1, t2, b.x * fast_rcp(den)) - b.w;
    return fast_tanh(e);
}

__global__ __launch_bounds__(256) void expr_kernel(const float* __restrict__ in,
                                                   float* __restrict__ out,
                                                   int n_rows) {
    const long long tid      = (long long)blockIdx.x * blockDim.x + threadIdx.x;
    const long long base_row = tid * 4;

    if (base_row + 3 < (long long)n_rows) {
        // Fast path: 4 full rows -> 8x global_load_b128 (NT), 1x global_store_b128 (NT)
        v4f res;
#pragma unroll
        for (int i = 0; i < 4; ++i) {
            const v4f* p = (const v4f*)(in + (base_row + i) * 8);
            v4f a = __builtin_nontemporal_load(p);
            v4f b = __builtin_nontemporal_load(p + 1);
            res[i] = eval_row(a, b);
        }
        __builtin_nontemporal_store(res, (v4f*)(out + base_row));
    } else if (base_row < (long long)n_rows) {
        // Tail (never taken for n_rows % 4 == 0, kept for generality).
        for (long long row = base_row; row < (long long)n_rows; ++row) {
            const v4f* p = (const v4f*)(in + row * 8);
            v4f a = __builtin_nontemporal_load(p);
            v4f b = __builtin_nontemporal_load(p + 1);
            float v = eval_row(a, b);
            __builtin_nontemporal_store(v, out + row);
        }
    }
}

extern "C" void kernel_launch(void* const* d_in, const int* in_sizes, int n_in,
                              void* d_out, int out_size, void* d_ws, size_t ws_size,
                              hipStream_t stream) {
    const float* in = (const float*)d_in[0];
    float* out      = (float*)d_out;
    const int n_rows = out_size;               // 8,388,608

    const int threads       = 256;             // 8 wave32s per block
    const int rows_per_thr  = 4;
    const long long n_thr   = ((long long)n_rows + rows_per_thr - 1) / rows_per_thr;
    const int blocks        = (int)((n_thr + threads - 1) / threads);   // 8192

    expr_kernel<<<blocks, threads, 0, stream>>>(in, out, n_rows);
}